// GraphModule_76630806495260
// MI455X (gfx1250) — compile-verified
//
#include <hip/hip_runtime.h>

#define N_NODES 100000

typedef __attribute__((ext_vector_type(2))) float v2f;
typedef __attribute__((ext_vector_type(8))) float v8f;

// ---------------- degree / normalization ----------------

__global__ void k_deg_init(float* __restrict__ deg, int n) {
    int i = blockIdx.x * blockDim.x + threadIdx.x;
    if (i < n) deg[i] = 1.0f;   // self-loop contributes 1
}

__global__ void k_deg_accum(const long long* __restrict__ dst, float* __restrict__ deg, int nE) {
    int e = blockIdx.x * blockDim.x + threadIdx.x;
    if (e < nE) atomicAdd(&deg[dst[e]], 1.0f);
}

__global__ void k_deg_to_dis(float* __restrict__ deg, int n) {
    int i = blockIdx.x * blockDim.x + threadIdx.x;
    if (i < n) deg[i] = rsqrtf(deg[i]);   // deg >= 1 always (self-loops)
}

// ---------------- WMMA fp32 GEMM: D[M x COUT] = act(A[M x K]) @ W[K x COUT] ----------------
// One wave per 16x16 output tile. blockDim.x = 32 * (COUT/16); block handles a
// 16-row strip of A across all COUT/16 column tiles. M (=100000) is a multiple of 16.
//
// V_WMMA_F32_16X16X4_F32 fragment layout (wave32):
//   A (16x4):  lanes 0-15 -> M=lane, v0=K(k0), v1=K(k0+1); lanes 16-31 -> v0=K(k0+2), v1=K(k0+3)
//   B (4x16):  lanes 0-15 -> N=lane, v0=K(k0), v1=K(k0+1); lanes 16-31 -> v0=K(k0+2), v1=K(k0+3)
//   C/D (16x16): VGPR r, lane -> M = r + 8*(lane>=16), N = lane&15

template <int K, int COUT, bool FUSE_BIAS_RELU>
__global__ void k_gemm_wmma(const float* __restrict__ A,
                            const float* __restrict__ W,
                            const float* __restrict__ bias,   // length K, applied to A before ReLU
                            float* __restrict__ D) {
    const int wave  = threadIdx.x >> 5;
    const int lane  = threadIdx.x & 31;
    const int lhalf = lane >> 4;     // 0 or 1
    const int lmod  = lane & 15;
    const int m0    = blockIdx.x * 16;
    const int n0    = wave * 16;

    const float* arow = A + (long)(m0 + lmod) * K;

    v8f acc = {};
#pragma unroll
    for (int k0 = 0; k0 < K; k0 += 4) {
        const int ka = k0 + 2 * lhalf;

        float a0 = arow[ka];
        float a1 = arow[ka + 1];
        if (FUSE_BIAS_RELU) {
            a0 = fmaxf(a0 + bias[ka], 0.0f);
            a1 = fmaxf(a1 + bias[ka + 1], 0.0f);
        }
        v2f a; a.x = a0; a.y = a1;

        v2f b;
        b.x = W[(long)ka * COUT + n0 + lmod];
        b.y = W[(long)(ka + 1) * COUT + n0 + lmod];

        // 8 args: (neg_a, A, neg_b, B, c_mod, C, reuse_a, reuse_b)
        acc = __builtin_amdgcn_wmma_f32_16x16x4_f32(
            false, a, false, b, (short)0, acc, false, false);
    }

#pragma unroll
    for (int r = 0; r < 8; ++r) {
        const int m = r + 8 * lhalf;
        D[(long)(m0 + m) * COUT + n0 + lmod] = acc[r];
    }
}

// ---------------- self-loop init: out[i,:] = H[i,:] * dis[i]^2 ----------------

template <int C>  // C multiple of 4; processes float4 per thread
__global__ void k_self_init(const float* __restrict__ H,
                            const float* __restrict__ dis,
                            float* __restrict__ out, int n) {
    const int C4 = C / 4;
    long gid = (long)blockIdx.x * blockDim.x + threadIdx.x;
    if (gid >= (long)n * C4) return;
    int i = (int)(gid / C4);
    float s = dis[i];
    s = s * s;
    float4 h = ((const float4*)H)[gid];
    float4 o;
    o.x = h.x * s; o.y = h.y * s; o.z = h.z * s; o.w = h.w * s;
    ((float4*)out)[gid] = o;
}

// ---------------- edge scatter: out[dst,:] += H[src,:] * dis[src]*dis[dst] ----------------
// Consecutive lanes = consecutive channel-quads of one edge -> coalesced float4 gathers.

template <int C, int LOGC4>  // C = 4 << LOGC4
__global__ void k_edge_scatter(const long long* __restrict__ src,
                               const long long* __restrict__ dst,
                               const float* __restrict__ dis,
                               const float* __restrict__ H,
                               float* __restrict__ out, int nE) {
    long gid = (long)blockIdx.x * blockDim.x + threadIdx.x;
    long total = (long)nE << LOGC4;
    if (gid >= total) return;
    int e = (int)(gid >> LOGC4);
    int c = (int)(gid & ((1 << LOGC4) - 1)) << 2;

    long s = src[e];
    long d = dst[e];
    float w = dis[s] * dis[d];

    float4 h = *(const float4*)(H + s * C + c);
    float* o = out + d * C + c;
    atomicAdd(o + 0, h.x * w);
    atomicAdd(o + 1, h.y * w);
    atomicAdd(o + 2, h.z * w);
    atomicAdd(o + 3, h.w * w);
}

// ---------------- LayerNorm over 64 channels, in place: out = LN(out + b2)*gamma + beta ----------------
// One wave per node; each lane owns channels {lane, lane+32}.

__global__ void k_layernorm64(float* __restrict__ out,
                              const float* __restrict__ b2,
                              const float* __restrict__ gamma,
                              const float* __restrict__ beta, int n) {
    const int wave = threadIdx.x >> 5;
    const int lane = threadIdx.x & 31;
    const int i = blockIdx.x * (blockDim.x >> 5) + wave;
    if (i >= n) return;

    float* row = out + (long)i * 64;
    float v0 = row[lane]      + b2[lane];
    float v1 = row[lane + 32] + b2[lane + 32];

    float sum = v0 + v1;
#pragma unroll
    for (int off = 16; off > 0; off >>= 1) sum += __shfl_xor(sum, off, 32);
    const float mu = sum * (1.0f / 64.0f);

    const float d0 = v0 - mu, d1 = v1 - mu;
    float ss = d0 * d0 + d1 * d1;
#pragma unroll
    for (int off = 16; off > 0; off >>= 1) ss += __shfl_xor(ss, off, 32);
    const float inv = rsqrtf(ss * (1.0f / 64.0f) + 1e-5f);

    row[lane]      = d0 * inv * gamma[lane]      + beta[lane];
    row[lane + 32] = d1 * inv * gamma[lane + 32] + beta[lane + 32];
}

// ---------------- launch ----------------

extern "C" void kernel_launch(void* const* d_in, const int* in_sizes, int n_in,
                              void* d_out, int out_size, void* d_ws, size_t ws_size,
                              hipStream_t stream) {
    const float*     x     = (const float*)d_in[0];
    const long long* ei    = (const long long*)d_in[1];   // int64 [2, E]
    const float*     W1    = (const float*)d_in[2];       // [64,128]
    const float*     b1    = (const float*)d_in[3];       // [128]
    const float*     W2    = (const float*)d_in[4];       // [128,64]
    const float*     b2    = (const float*)d_in[5];       // [64]
    const float*     gamma = (const float*)d_in[6];       // [64]
    const float*     beta  = (const float*)d_in[7];       // [64]

    const int nE = in_sizes[1] / 2;
    const long long* srcIdx = ei;
    const long long* dstIdx = ei + nE;

    float* out  = (float*)d_out;                          // [N, 64]
    float* ws   = (float*)d_ws;
    float* dis  = ws;                                     // [N]  (deg, then rsqrt(deg))
    float* bufA = ws + N_NODES;                           // [N,128] h1, later h2
    float* bufB = bufA + (size_t)N_NODES * 128;           // [N,128] agg1

    const int T = 256;

    // 1) degrees -> dis = rsqrt(deg)
    k_deg_init  <<<(N_NODES + T - 1) / T, T, 0, stream>>>(dis, N_NODES);
    k_deg_accum <<<(nE + T - 1) / T,      T, 0, stream>>>(dstIdx, dis, nE);
    k_deg_to_dis<<<(N_NODES + T - 1) / T, T, 0, stream>>>(dis, N_NODES);

    // 2) layer 1 transform: bufA = x @ W1   (WMMA fp32, 8 waves/block = 8 column tiles)
    k_gemm_wmma<64, 128, false><<<N_NODES / 16, 256, 0, stream>>>(x, W1, nullptr, bufA);

    // 3) layer 1 aggregation: bufB = scatter(bufA)  (bias+ReLU deferred into next GEMM)
    k_self_init<128><<<(int)(((long)N_NODES * 32 + T - 1) / T), T, 0, stream>>>(bufA, dis, bufB, N_NODES);
    k_edge_scatter<128, 5><<<(int)((((long)nE << 5) + T - 1) / T), T, 0, stream>>>(
        srcIdx, dstIdx, dis, bufA, bufB, nE);

    // 4) layer 2 transform: bufA = relu(bufB + b1) @ W2   (4 waves/block)
    k_gemm_wmma<128, 64, true><<<N_NODES / 16, 128, 0, stream>>>(bufB, W2, b1, bufA);

    // 5) layer 2 aggregation directly into d_out
    k_self_init<64><<<(int)(((long)N_NODES * 16 + T - 1) / T), T, 0, stream>>>(bufA, dis, out, N_NODES);
    k_edge_scatter<64, 4><<<(int)((((long)nE << 4) + T - 1) / T), T, 0, stream>>>(
        srcIdx, dstIdx, dis, bufA, out, nE);

    // 6) in-place LayerNorm (adds b2 first), one wave per node
    k_layernorm64<<<(N_NODES + 7) / 8, 256, 0, stream>>>(out, b2, gamma, beta, N_NODES);
}